// BiLSTM_CRF_45011257262599
// MI455X (gfx1250) — compile-verified
//
#include <hip/hip_runtime.h>

typedef __attribute__((ext_vector_type(2))) float v2f;
typedef __attribute__((ext_vector_type(8))) float v8f;

#define TT 2048
#define EMBD 256
#define H2D 256
#define G4D 1024
#define CC 12
#define START_TAG 10
#define STOP_TAG 11
#define NEGV (-10000.0f)

__device__ __forceinline__ float sigmoidf(float x) { return 1.0f / (1.0f + expf(-x)); }

// ---------------------------------------------------------------------------
// Prep: gather embeddings X[t][e] = emb[sentence[t]][e]; repack Whh into a
// k-chunked layout W4[c][j][r] (c = k/4, r = k%4, j = gate index) so the
// serial recurrence can stream weights as fully-coalesced global_load_b128.
// Total elements: 2048*256 + 2*256*1024 = 1048576 = 4096 blocks * 256 threads.
// ---------------------------------------------------------------------------
__global__ void prep_kernel(const int* __restrict__ sentence, const float* __restrict__ emb,
                            const float* __restrict__ Whh_f, const float* __restrict__ Whh_b,
                            float* __restrict__ X, float* __restrict__ W4_f,
                            float* __restrict__ W4_b) {
  int id = blockIdx.x * blockDim.x + threadIdx.x;
  if (id < TT * EMBD) {
    int t = id >> 8, e = id & 255;
    X[id] = emb[(size_t)sentence[t] * EMBD + e];
  } else {
    int seg = id - TT * EMBD;
    const float* src = (seg < H2D * G4D) ? Whh_f : Whh_b;
    float* dst = (seg < H2D * G4D) ? W4_f : W4_b;
    int i = (seg < H2D * G4D) ? seg : seg - H2D * G4D;
    // i = c*4096 + j*4 + r ; k = c*4 + r
    int c = i >> 12;
    int rem = i & 4095;
    int j = rem >> 2;
    int r = rem & 3;
    int k = (c << 2) | r;
    dst[i] = src[j * H2D + k];
  }
}

// ---------------------------------------------------------------------------
// Gate pre-activations via fp32 WMMA:
//   G[t][j] = sum_k X[t][k] * Wih[j][k] + bih[j] + bhh[j]
// One wave per 16x16 tile of G; K=256 consumed in 64 steps of
// V_WMMA_F32_16X16X4_F32. Exact fp32 math (gates feed sigmoid/tanh).
// A layout (16x4 f32): lane<16 -> {A[m][kb],A[m][kb+1]}, lane>=16 -> k+2,k+3.
// B layout (4x16 f32): same pattern in N role; C/D: VGPR r = row r + 8*(lane/16).
// ---------------------------------------------------------------------------
__global__ void gates_wmma_kernel(const float* __restrict__ X,
                                  const float* __restrict__ Wih_f, const float* __restrict__ bih_f,
                                  const float* __restrict__ bhh_f,
                                  const float* __restrict__ Wih_b, const float* __restrict__ bih_b,
                                  const float* __restrict__ bhh_b,
                                  float* __restrict__ Gf, float* __restrict__ Gb) {
  const int lane = threadIdx.x;           // 0..31, one wave per block
  const int nt = blockIdx.x;              // 0..63  (N tiles over 1024 gates)
  const int mt = blockIdx.y;              // 0..127 (M tiles over 2048 timesteps)
  const int dir = blockIdx.z;             // 0=fwd, 1=bwd
  const float* W  = dir ? Wih_b : Wih_f;
  const float* bi = dir ? bih_b : bih_f;
  const float* bh = dir ? bhh_b : bhh_f;
  float* G = dir ? Gb : Gf;

  const int l15 = lane & 15;
  const int half = lane >> 4;             // k sub-offset selector
  const int m = mt * 16 + l15;            // A row (timestep)
  const int n = nt * 16 + l15;            // B column (gate index)
  const float* xrow = X + (size_t)m * EMBD;
  const float* wrow = W + (size_t)n * EMBD;

  v8f acc = {0.f, 0.f, 0.f, 0.f, 0.f, 0.f, 0.f, 0.f};
#pragma unroll 4
  for (int kb = 0; kb < EMBD; kb += 4) {
    const int k = kb + 2 * half;
    v2f a; a.x = xrow[k]; a.y = xrow[k + 1];
    v2f b; b.x = wrow[k]; b.y = wrow[k + 1];
    acc = __builtin_amdgcn_wmma_f32_16x16x4_f32(false, a, false, b, (short)0, acc,
                                                false, false);
  }
  const float bias = bi[n] + bh[n];
#pragma unroll
  for (int r = 0; r < 8; ++r) {
    const int mm = mt * 16 + r + 8 * half;  // D row for this VGPR
    G[(size_t)mm * G4D + n] = acc[r] + bias;
  }
}

// ---------------------------------------------------------------------------
// Serial LSTM recurrence: one WGP (1024 threads) per direction.
// Thread j computes gate pre-activation j each step: 64 iterations of
// (global_load_b128 weights, ds_load_b128 h, 4 FMAs). Weights are the
// k-chunked W4 layout: chunk c holds [j][0..3] for k = 4c..4c+3, so lane j
// reads 16B at W4 + c*16KB + j*16 -> consecutive lanes, consecutive 16B.
// The 1MB weight block stays L2-resident across all 2048 steps.
// Threads < 256 apply the cell update; c-state stays in registers.
// ---------------------------------------------------------------------------
__global__ void __launch_bounds__(1024) lstm_recur_kernel(
    const float* __restrict__ Gf, const float* __restrict__ Gb,
    const float* __restrict__ W4_f, const float* __restrict__ W4_b,
    const float* __restrict__ h0, const float* __restrict__ c0,
    float* __restrict__ hf, float* __restrict__ hb) {
  const int dir = blockIdx.x;
  const float* G  = dir ? Gb : Gf;
  const float* W4 = dir ? W4_b : W4_f;
  float* hs = dir ? hb : hf;

  __shared__ __align__(16) float h_sh[H2D];
  __shared__ float ga[G4D];

  const int tid = threadIdx.x;
  float c_reg = 0.0f;
  if (tid < H2D) {
    h_sh[tid] = h0[dir * H2D + tid];
    c_reg = c0[dir * H2D + tid];
  }
  __syncthreads();

  const float* wp = W4 + ((size_t)tid << 2);  // 16B per lane, lane-consecutive
  for (int s = 0; s < TT; ++s) {
    const int t = dir ? (TT - 1 - s) : s;
    float acc = G[(size_t)t * G4D + tid];
#pragma unroll 8
    for (int c = 0; c < H2D / 4; ++c) {
      const float4 w  = *(const float4*)(wp + ((size_t)c << 12));
      const float4 hv = *(const float4*)(h_sh + (c << 2));
      acc += hv.x * w.x + hv.y * w.y + hv.z * w.z + hv.w * w.w;
    }
    ga[tid] = acc;
    __syncthreads();
    if (tid < H2D) {
      // PyTorch gate order: i, f, g, o
      const float ig = sigmoidf(ga[tid]);
      const float fg = sigmoidf(ga[H2D + tid]);
      const float gg = tanhf(ga[2 * H2D + tid]);
      const float og = sigmoidf(ga[3 * H2D + tid]);
      c_reg = fg * c_reg + ig * gg;
      const float hn = og * tanhf(c_reg);
      h_sh[tid] = hn;
      hs[(size_t)t * H2D + tid] = hn;
    }
    __syncthreads();
  }
}

// ---------------------------------------------------------------------------
// feats[t][c] = [hf[t], hb[t]] . Wlin[c] + blin[c]   (one thread per output)
// ---------------------------------------------------------------------------
__global__ void feats_kernel(const float* __restrict__ hf, const float* __restrict__ hb,
                             const float* __restrict__ Wlin, const float* __restrict__ blin,
                             float* __restrict__ feats) {
  const int id = blockIdx.x * blockDim.x + threadIdx.x;
  if (id >= TT * CC) return;
  const int t = id / CC, c = id % CC;
  float acc = blin[c];
  const float4* w = (const float4*)(Wlin + (size_t)c * 512);
  const float4* a = (const float4*)(hf + (size_t)t * H2D);
  const float4* b = (const float4*)(hb + (size_t)t * H2D);
#pragma unroll 4
  for (int k = 0; k < H2D / 4; ++k) {
    const float4 av = a[k], wv = w[k];
    acc += av.x * wv.x + av.y * wv.y + av.z * wv.z + av.w * wv.w;
  }
#pragma unroll 4
  for (int k = 0; k < H2D / 4; ++k) {
    const float4 bv = b[k], wv = w[64 + k];
    acc += bv.x * wv.x + bv.y * wv.y + bv.z * wv.z + bv.w * wv.w;
  }
  feats[id] = acc;
}

// ---------------------------------------------------------------------------
// Viterbi + backtrace in a single wave. Lane = destination tag (12 active).
// Strict-greater max keeps the first index, matching jnp.argmax.
// out[0] = path score, out[1..T] = path tags (as float).
// ---------------------------------------------------------------------------
__global__ void viterbi_kernel(const float* __restrict__ feats, const float* __restrict__ trans,
                               int* __restrict__ bp_g, float* __restrict__ out) {
  __shared__ float fv[CC];
  __shared__ float ts[CC * CC];
  const int tid = threadIdx.x;
  if (tid < CC) {
    for (int p = 0; p < CC; ++p) ts[tid * CC + p] = trans[tid * CC + p];
    fv[tid] = (tid == START_TAG) ? 0.0f : NEGV;
  }
  __syncthreads();

  for (int t = 0; t < TT; ++t) {
    float best = -3.4e38f;
    int bprev = 0;
    if (tid < CC) {
#pragma unroll
      for (int p = 0; p < CC; ++p) {
        const float s = fv[p] + ts[tid * CC + p];
        if (s > best) { best = s; bprev = p; }
      }
    }
    __syncthreads();
    if (tid < CC) {
      bp_g[t * CC + tid] = bprev;
      fv[tid] = best + feats[t * CC + tid];
    }
    __syncthreads();
  }

  if (tid == 0) {
    float bs = -3.4e38f;
    int bt = 0;
    for (int p = 0; p < CC; ++p) {
      const float s = fv[p] + ts[STOP_TAG * CC + p];
      if (s > bs) { bs = s; bt = p; }
    }
    out[0] = bs;
    int tag = bt;
    for (int t = TT - 1; t >= 0; --t) {
      out[1 + t] = (float)tag;
      tag = bp_g[t * CC + tag];
    }
  }
}

// ---------------------------------------------------------------------------
extern "C" void kernel_launch(void* const* d_in, const int* in_sizes, int n_in,
                              void* d_out, int out_size, void* d_ws, size_t ws_size,
                              hipStream_t stream) {
  (void)in_sizes; (void)n_in; (void)out_size; (void)ws_size;
  const int*   sentence = (const int*)d_in[0];
  const float* emb      = (const float*)d_in[1];
  const float* Wih_f    = (const float*)d_in[2];
  const float* Whh_f    = (const float*)d_in[3];
  const float* bih_f    = (const float*)d_in[4];
  const float* bhh_f    = (const float*)d_in[5];
  const float* Wih_b    = (const float*)d_in[6];
  const float* Whh_b    = (const float*)d_in[7];
  const float* bih_b    = (const float*)d_in[8];
  const float* bhh_b    = (const float*)d_in[9];
  const float* Wlin     = (const float*)d_in[10];
  const float* blin     = (const float*)d_in[11];
  const float* h0       = (const float*)d_in[12];
  const float* c0       = (const float*)d_in[13];
  const float* trans    = (const float*)d_in[14];
  float* out = (float*)d_out;

  // Workspace layout (float offsets)
  float* ws    = (float*)d_ws;
  float* X     = ws;                   // 2048*256        =  524288
  float* W4_f  = ws + 524288;          // 256*1024        =  262144
  float* W4_b  = ws + 786432;          // 256*1024        =  262144
  float* Gf    = ws + 1048576;         // 2048*1024       = 2097152
  float* Gb    = ws + 3145728;         // 2048*1024       = 2097152
  float* hf    = ws + 5242880;         // 2048*256        =  524288
  float* hb    = ws + 5767168;         // 2048*256        =  524288
  float* feats = ws + 6291456;         // 2048*12         =   24576
  int*   bp    = (int*)(ws + 6316032); // 2048*12 ints    =   24576

  prep_kernel<<<4096, 256, 0, stream>>>(sentence, emb, Whh_f, Whh_b, X, W4_f, W4_b);
  gates_wmma_kernel<<<dim3(64, 128, 2), 32, 0, stream>>>(X, Wih_f, bih_f, bhh_f,
                                                         Wih_b, bih_b, bhh_b, Gf, Gb);
  lstm_recur_kernel<<<2, 1024, 0, stream>>>(Gf, Gb, W4_f, W4_b, h0, c0, hf, hb);
  feats_kernel<<<96, 256, 0, stream>>>(hf, hb, Wlin, blin, feats);
  viterbi_kernel<<<1, 32, 0, stream>>>(feats, trans, bp, out);
}